// MAC_23871428231275
// MI455X (gfx1250) — compile-verified
//
#include <hip/hip_runtime.h>
#include <hip/hip_bf16.h>
#include <math.h>

typedef __attribute__((ext_vector_type(2))) float v2f;
typedef __attribute__((ext_vector_type(8))) float v8f;

#define N_AGENTS 64
#define HID      1024
#define COMM_DIM 256
#define VOCAB    4096
#define VSPLIT   32
#define VCHUNK   (VOCAB / VSPLIT)   // 128

#define FLT_BIG  3.402823466e38f

// ---------------------------------------------------------------------------
// Kernel 1: sample = eps * exp(0.5*(h@W_var + b_var)) + (h@W_mu + b_mu)
// One wave (32 lanes) per 16x16 output tile; WMMA f32 16x16x4 over K=1024.
// Also initializes token[] tile to +MAX for the atomic-min pass (same tile
// coordinates, re-done every call so graph replays stay deterministic).
// ---------------------------------------------------------------------------
__global__ __launch_bounds__(32) void vib_gemm_kernel(
    const float* __restrict__ h,    const float* __restrict__ Wmu,
    const float* __restrict__ bmu,  const float* __restrict__ Wvar,
    const float* __restrict__ bvar, const float* __restrict__ eps,
    float* __restrict__ sample,     float* __restrict__ token)
{
  const int lane = threadIdx.x;       // 0..31
  const int tile = blockIdx.x;        // 0..63
  const int mt   = tile >> 4;         // 0..3   (M tile)
  const int nt   = tile & 15;         // 0..15  (N tile)
  const int half = lane >> 4;         // 0 or 1 (K-pair select)
  const int l16  = lane & 15;

  const int mrow = mt * 16 + l16;     // A row owned by this lane
  const int ncol = nt * 16 + l16;     // B/C column owned by this lane

  const float* __restrict__ hrow = h + mrow * HID;

  v8f cmu = {};
  v8f cvar = {};

  #pragma unroll 4
  for (int k = 0; k < HID; k += 4) {
    const int ka = k + half * 2;
    v2f a;
    a.x = hrow[ka + 0];
    a.y = hrow[ka + 1];

    v2f bm, bv;
    bm.x = Wmu [(ka + 0) * COMM_DIM + ncol];
    bm.y = Wmu [(ka + 1) * COMM_DIM + ncol];
    bv.x = Wvar[(ka + 0) * COMM_DIM + ncol];
    bv.y = Wvar[(ka + 1) * COMM_DIM + ncol];

    // 8 args: (neg_a, A, neg_b, B, c_mod, C, reuse_a, reuse_b)
    cmu  = __builtin_amdgcn_wmma_f32_16x16x4_f32(false, a, false, bm,
                                                 (short)0, cmu,  false, false);
    cvar = __builtin_amdgcn_wmma_f32_16x16x4_f32(false, a, false, bv,
                                                 (short)0, cvar, false, false);
  }

  const float biasm = bmu[ncol];
  const float biasv = bvar[ncol];

  #pragma unroll
  for (int r = 0; r < 8; ++r) {
    const int m  = mt * 16 + r + half * 8;
    const float mu = cmu[r]  + biasm;
    const float lv = cvar[r] + biasv;
    sample[m * COMM_DIM + ncol] = eps[m * COMM_DIM + ncol] * __expf(0.5f * lv) + mu;
    token [m * COMM_DIM + ncol] = FLT_BIG;   // init for atomic-min pass
  }
}

// ---------------------------------------------------------------------------
// Kernel 2: partial min over a vocab chunk, merged via atomic fmin.
// Grid: (4 d-chunks) x (32 v-chunks) = 128 blocks -> spreads the dominant
// 200M-lane-op scan across the WGPs instead of 4.
// Block = 256 threads = 4 agent-groups x 64 dims; 16 agents in registers so
// each vocab scalar load feeds 16 sub/mul/min triples.
// ---------------------------------------------------------------------------
__global__ __launch_bounds__(256) void vq_partial_kernel(
    const float* __restrict__ sample, const float* __restrict__ vocab,
    float* __restrict__ token)
{
  const int t  = threadIdx.x;
  const int dl = t & 63;                     // local dim 0..63
  const int g  = t >> 6;                     // agent group 0..3
  const int d  = blockIdx.x * 64 + dl;       // global dim
  const int a0 = g * 16;
  const int v0 = blockIdx.y * VCHUNK;

  float sv[16];
  #pragma unroll
  for (int i = 0; i < 16; ++i) sv[i] = sample[(a0 + i) * COMM_DIM + d];

  float mn[16];
  #pragma unroll
  for (int i = 0; i < 16; ++i) mn[i] = FLT_BIG;

  const float* __restrict__ vc = vocab + v0 * COMM_DIM + d;
  for (int v = 0; v < VCHUNK; v += 4) {
    const float c0 = vc[(v + 0) * COMM_DIM];
    const float c1 = vc[(v + 1) * COMM_DIM];
    const float c2 = vc[(v + 2) * COMM_DIM];
    const float c3 = vc[(v + 3) * COMM_DIM];
    #pragma unroll
    for (int i = 0; i < 16; ++i) {
      float d0 = sv[i] - c0; d0 *= d0;
      float d1 = sv[i] - c1; d1 *= d1;
      float d2 = sv[i] - c2; d2 *= d2;
      float d3 = sv[i] - c3; d3 *= d3;
      const float m01 = fminf(d0, d1);
      const float m23 = fminf(d2, d3);
      mn[i] = fminf(mn[i], fminf(m01, m23));
    }
  }

  #pragma unroll
  for (int i = 0; i < 16; ++i) {
    // order-independent reduction -> deterministic across replays
    __hip_atomic_fetch_min(&token[(a0 + i) * COMM_DIM + d], mn[i],
                           __ATOMIC_RELAXED, __HIP_MEMORY_SCOPE_AGENT);
  }
}

// ---------------------------------------------------------------------------
// Kernel 3: comm_out[j,d] = (sum_i token[i,d]) - token[j,d]   (mask = 1-eye)
// 4 blocks x 256 threads; block owns all 64 agents for its 64-dim slice.
// ---------------------------------------------------------------------------
__global__ __launch_bounds__(256) void comm_kernel(
    const float* __restrict__ token, float* __restrict__ comm_out)
{
  __shared__ float tok[N_AGENTS][64];   // [agent][local dim]
  __shared__ float colsum[64];

  const int t  = threadIdx.x;
  const int dl = t & 63;
  const int g  = t >> 6;
  const int d  = blockIdx.x * 64 + dl;
  const int a0 = g * 16;

  #pragma unroll
  for (int i = 0; i < 16; ++i) tok[a0 + i][dl] = token[(a0 + i) * COMM_DIM + d];
  __syncthreads();

  if (t < 64) {
    float s = 0.f;
    #pragma unroll 8
    for (int a = 0; a < N_AGENTS; ++a) s += tok[a][t];
    colsum[t] = s;
  }
  __syncthreads();

  const float cs = colsum[dl];
  #pragma unroll
  for (int i = 0; i < 16; ++i) {
    const int j = a0 + i;
    comm_out[j * COMM_DIM + d] = cs - tok[j][dl];
  }
}

extern "C" void kernel_launch(void* const* d_in, const int* in_sizes, int n_in,
                              void* d_out, int out_size, void* d_ws, size_t ws_size,
                              hipStream_t stream) {
  (void)in_sizes; (void)n_in; (void)out_size; (void)ws_size;
  const float* h     = (const float*)d_in[0];
  const float* Wmu   = (const float*)d_in[1];
  const float* bmu   = (const float*)d_in[2];
  const float* Wvar  = (const float*)d_in[3];
  const float* bvar  = (const float*)d_in[4];
  const float* vocab = (const float*)d_in[5];
  const float* eps   = (const float*)d_in[6];
  float* out    = (float*)d_out;

  float* sample = (float*)d_ws;                       // 64*256 floats = 64 KB
  float* token  = sample + N_AGENTS * COMM_DIM;       // 64*256 floats = 64 KB

  vib_gemm_kernel<<<64, 32, 0, stream>>>(h, Wmu, bmu, Wvar, bvar, eps,
                                         sample, token);
  vq_partial_kernel<<<dim3(4, VSPLIT), 256, 0, stream>>>(sample, vocab, token);
  comm_kernel<<<4, 256, 0, stream>>>(token, out);
}